// block_dtw_46651934769385
// MI455X (gfx1250) — compile-verified
//
#include <hip/hip_runtime.h>

typedef __attribute__((ext_vector_type(2))) float v2f;
typedef __attribute__((ext_vector_type(8))) float v8f;

#define BLK  128
#define CDIM 64
#define INF_ 1e30f

// One workgroup (128 threads = 4 waves) per soft-DTW block problem.
__global__ __launch_bounds__(128) void sdtw_kernel(const float* __restrict__ X,
                                                   const float* __restrict__ Y,
                                                   float* __restrict__ ws) {
  __shared__ float Ds[BLK * BLK];     // 64 KB cost matrix
  __shared__ float x2s[BLK];
  __shared__ float y2s[BLK];
  __shared__ float Rl[2][BLK];        // rolling anti-diagonal buffers

  const int tid = threadIdx.x;
  const int p   = blockIdx.x;
  const float* Xp = X + (size_t)p * BLK * CDIM;
  const float* Yp = Y + (size_t)p * BLK * CDIM;

  // ---- Phase 1: squared norms (thread i owns row i) ----
  {
    const float4* xr = (const float4*)(Xp + tid * CDIM);
    const float4* yr = (const float4*)(Yp + tid * CDIM);
    float sx = 0.f, sy = 0.f;
#pragma unroll
    for (int c = 0; c < CDIM / 4; ++c) {
      float4 a = xr[c]; float4 b = yr[c];
      sx += a.x*a.x + a.y*a.y + a.z*a.z + a.w*a.w;
      sy += b.x*b.x + b.y*b.y + b.z*b.z + b.w*b.w;
    }
    x2s[tid] = sx; y2s[tid] = sy;
  }
  __syncthreads();

  // ---- Phase 2: D = x2 + y2 - 2*X*Y^T via V_WMMA_F32_16X16X4_F32 ----
  // A-fragment layout (ISA 7.12.2, 32-bit A 16x4): lanes 0-15: M=lane, K=kb*4+{0,1};
  // lanes 16-31: M=lane-16, K=kb*4+{2,3}.  B mirrors from Y rows (B = Y^T).
  {
    const int w    = tid >> 5;
    const int lane = tid & 31;
    const int half = lane >> 4;
    const int lm   = lane & 15;

    v2f A[2][16];
#pragma unroll
    for (int mb = 0; mb < 2; ++mb) {
      const int M0 = (2 * w + mb) * 16;
      const float* base = Xp + (size_t)(M0 + lm) * CDIM + half * 2;
#pragma unroll
      for (int kb = 0; kb < 16; ++kb)
        A[mb][kb] = *(const v2f*)(base + kb * 4);
    }

#pragma unroll 1
    for (int nb = 0; nb < 8; ++nb) {
      const int N0 = nb * 16;
      v2f Bf[16];
      const float* base = Yp + (size_t)(N0 + lm) * CDIM + half * 2;
#pragma unroll
      for (int kb = 0; kb < 16; ++kb)
        Bf[kb] = *(const v2f*)(base + kb * 4);
      const float y2v = y2s[N0 + lm];

#pragma unroll
      for (int mb = 0; mb < 2; ++mb) {
        const int M0 = (2 * w + mb) * 16;
        v8f acc = {};
#pragma unroll
        for (int kb = 0; kb < 16; ++kb)
          acc = __builtin_amdgcn_wmma_f32_16x16x4_f32(
              /*neg_a=*/false, A[mb][kb], /*neg_b=*/false, Bf[kb],
              /*c_mod=*/(short)0, acc, /*reuse_a=*/false, /*reuse_b=*/false);
        // C/D layout: VGPR r, lanes 0-15 -> M=M0+r, lanes 16-31 -> M=M0+r+8; N = N0+lane%16
#pragma unroll
        for (int r = 0; r < 8; ++r) {
          const int m = M0 + r + half * 8;
          Ds[m * BLK + N0 + lm] = fmaf(-2.f, acc[r], x2s[m] + y2v);
        }
      }
    }
  }

  // init rolling diagonal buffers
  Rl[0][tid] = INF_;
  Rl[1][tid] = INF_;
  __syncthreads();

  // ---- Phase 3: anti-diagonal wavefront, softmin (gamma = 1) ----
  const int i = tid;
  float myPrev = INF_;   // R[i] on diagonal d-1 (own row, kept in register)
  float nm2    = INF_;   // R[i-1] on diagonal d-2 (carried: last step's nm1)
  for (int d = 0; d < 2 * BLK - 1; ++d) {
    const float nm1 = (i == 0) ? INF_ : Rl[(d + 1) & 1][i - 1]; // R[i-1] @ d-1
    const float a = myPrev;                                      // R[i, j-1]
    const float b = nm1;                                         // R[i-1, j]
    const float c = (i == 0) ? ((d == 0) ? 0.f : INF_) : nm2;    // R[i-1, j-1]
    const float m = fminf(a, fminf(b, c));
    const float s = __expf(m - a) + __expf(m - b) + __expf(m - c);
    const float smin = m - __logf(s);
    const int  j     = d - i;
    const bool valid = (j >= 0) && (j < BLK);
    const int  jc    = valid ? j : 0;
    const float rnew = valid ? (Ds[i * BLK + jc] + smin) : INF_;
    Rl[d & 1][i] = rnew;
    nm2    = nm1;
    myPrev = rnew;
    __syncthreads();
  }
  if (i == BLK - 1) ws[p] = myPrev;   // R[127,127]
}

// Ordered, deterministic per-batch reduction: out[b] = sum_k ws[b*nb + k]
__global__ void reduce_kernel(const float* __restrict__ ws, float* __restrict__ out,
                              int nb, int nbatch) {
  const int t = threadIdx.x + blockIdx.x * blockDim.x;
  if (t < nbatch) {
    float s = 0.f;
    for (int k = 0; k < nb; ++k) s += ws[t * nb + k];
    out[t] = s;
  }
}

extern "C" void kernel_launch(void* const* d_in, const int* in_sizes, int n_in,
                              void* d_out, int out_size, void* d_ws, size_t ws_size,
                              hipStream_t stream) {
  const float* x  = (const float*)d_in[0];
  const float* xr = (const float*)d_in[1];
  float* out = (float*)d_out;
  float* ws  = (float*)d_ws;

  const int P  = in_sizes[0] / (BLK * CDIM);   // total block problems (1024)
  const int nb = P / out_size;                 // blocks per batch (32)

  sdtw_kernel<<<P, BLK, 0, stream>>>(x, xr, ws);
  reduce_kernel<<<1, (out_size + 31) / 32 * 32, 0, stream>>>(ws, out, nb, out_size);
}